// SelectivePolarityLoss_22247930593475
// MI455X (gfx1250) — compile-verified
//
#include <hip/hip_runtime.h>
#include <cstdint>

// Problem constants (from reference): VOCAB=200000, D=128, P=100000, K=8
#define DIM        128
#define KDIMS      8
#define WAVESZ     32
#define WAVES_PB   8
#define PPW        4                      // pairs per wave (double-buffered pipeline)
#define BLOCK      (WAVESZ * WAVES_PB)    // 256 threads = 8 wave32

static constexpr float POLARITY_WEIGHT   = 1.0f;
static constexpr float SIMILARITY_WEIGHT = 0.5f;

typedef int v4i __attribute__((vector_size(16)));
typedef __attribute__((address_space(1))) v4i *gas_v4i;  // global (AS1) v4i*
typedef __attribute__((address_space(3))) v4i *las_v4i;  // LDS    (AS3) v4i*

__device__ __forceinline__ void async_b128_to_lds(const float* g, float* l) {
#if defined(__AMDGCN__) && __has_builtin(__builtin_amdgcn_global_load_async_to_lds_b128)
  __builtin_amdgcn_global_load_async_to_lds_b128(
      (gas_v4i)(v4i*)(float*)g, (las_v4i)(v4i*)l, /*offset=*/0, /*cpol=*/0);
#else
  *(float4*)l = *(const float4*)g;   // fallback: direct b128 load + ds store
#endif
}

template <int N>
__device__ __forceinline__ void wait_async_le() {
#if defined(__AMDGCN__) && __has_builtin(__builtin_amdgcn_s_wait_asynccnt)
  __builtin_amdgcn_s_wait_asynccnt(N);
#elif defined(__AMDGCN__) && __has_builtin(__builtin_amdgcn_global_load_async_to_lds_b128)
  asm volatile("s_wait_asynccnt %0" :: "i"(N) : "memory");
#endif
}

__global__ __launch_bounds__(BLOCK) void pair_loss_kernel(
    const float* __restrict__ emb,        // [VOCAB, 128]
    const int*   __restrict__ pair_ids,   // [P, 2]
    const int*   __restrict__ dims,       // [P, 8]
    float*       __restrict__ partials,   // [gridDim.x]
    int npairs)
{
  // Per-wave double buffer: [wave][buf][v1(128) | v2(128)]  -> 16 KB/block
  __shared__ float rows[WAVES_PB][2][2 * DIM];
  __shared__ float wsum[WAVES_PB];

  const int lane = threadIdx.x & (WAVESZ - 1);
  const int wave = threadIdx.x >> 5;
  const int base = (blockIdx.x * WAVES_PB + wave) * PPW;

  int n = npairs - base;
  n = (n < 0) ? 0 : ((n > PPW) ? PPW : n);

  // Issue both 512B rows of a pair: 32 lanes x 16B async b128 each.
  auto issue = [&](int buf, int pair) {
    const int id1 = __builtin_amdgcn_readfirstlane(pair_ids[2 * pair + 0]);
    const int id2 = __builtin_amdgcn_readfirstlane(pair_ids[2 * pair + 1]);
    async_b128_to_lds(emb + (size_t)id1 * DIM + lane * 4, &rows[wave][buf][lane * 4]);
    async_b128_to_lds(emb + (size_t)id2 * DIM + lane * 4, &rows[wave][buf][DIM + lane * 4]);
  };

  float acc = 0.0f;
  if (n > 0) issue(0, base);              // prime the pipeline

  for (int i = 0; i < n; ++i) {
    const int pair = base + i;
    const bool has_next = (i + 1 < n);
    if (has_next) issue((i + 1) & 1, pair + 1);   // prefetch next pair's rows

    // Overlap with the copy: selected-dim index + dedup (no LDS data needed).
    const int myk = lane & (KDIMS - 1);
    const int dk  = dims[pair * KDIMS + myk];
    bool dup = false;
#pragma unroll
    for (int j = 0; j < KDIMS - 1; ++j) {
      const int dj = __shfl(dk, j, WAVESZ);
      if (j < myk && dj == dk) dup = true;
    }
    // unique masked-dim count via ballot (lanes 0..7 carry the 8 entries)
    const int ucnt =
        __popcll(__ballot((lane < KDIMS) && !dup));

    // Async loads complete in order: <=2 outstanding  ==> current buffer done.
    if (has_next) wait_async_le<2>(); else wait_async_le<0>();

    const float* v1 = rows[wave][i & 1];
    const float* v2 = rows[wave][i & 1] + DIM;

    // Full sum of (v1-v2)^2 over all 128 dims; 4 per lane.
    float total = 0.0f;
#pragma unroll
    for (int j = 0; j < 4; ++j) {
      const float dlt = v1[lane * 4 + j] - v2[lane * 4 + j];
      total += dlt * dlt;
    }

    // Lanes 0..7: polarity term for entry k (duplicates included) and the
    // unique-dim correction to turn the full sum into the masked sum.
    float per = 0.0f, corr = 0.0f;
    if (lane < KDIMS) {
      const float a1 = v1[dk];
      const float a2 = v2[dk];
      const float s1 = (a1 > 0.0f) ? 1.0f : ((a1 < 0.0f) ? -1.0f : 0.0f);
      const float s2 = (a2 > 0.0f) ? 1.0f : ((a2 < 0.0f) ? -1.0f : 0.0f);
      const float sp = s1 * s2;
      const float ab = fabsf(a1) + fabsf(a2);
      per = (sp > 0.0f) ? ab : ((sp < 0.0f) ? -0.5f * ab : 0.1f);
      if (!dup) {
        const float dd = a1 - a2;
        corr = dd * dd;
      }
    }

    // Butterfly reductions across the wave32 (3 float chains).
#pragma unroll
    for (int off = 16; off >= 1; off >>= 1) {
      total += __shfl_xor(total, off, WAVESZ);
      per   += __shfl_xor(per,   off, WAVESZ);
      corr  += __shfl_xor(corr,  off, WAVESZ);
    }

    // cnt = 128 - unique >= 120 > 0 always, so no cnt==0 branch needed.
    const float cnt = (float)(DIM - ucnt);
    const float sim = (total - corr) / cnt;
    acc += POLARITY_WEIGHT * per + SIMILARITY_WEIGHT * sim;
  }

  if (lane == 0) wsum[wave] = acc;
  __syncthreads();
  if (threadIdx.x == 0) {
    float s = 0.0f;
#pragma unroll
    for (int w = 0; w < WAVES_PB; ++w) s += wsum[w];
    partials[blockIdx.x] = s;
  }
}

// Deterministic single-block reduction (fixed traversal order every call).
__global__ __launch_bounds__(256) void final_reduce_kernel(
    const float* __restrict__ partials, int n, float* __restrict__ out, float scale)
{
  __shared__ float sm[256];
  float s = 0.0f;
  for (int i = threadIdx.x; i < n; i += 256) s += partials[i];
  sm[threadIdx.x] = s;
  __syncthreads();
  for (int off = 128; off > 0; off >>= 1) {
    if ((int)threadIdx.x < off) sm[threadIdx.x] += sm[threadIdx.x + off];
    __syncthreads();
  }
  if (threadIdx.x == 0) out[0] = sm[0] * scale;
}

extern "C" void kernel_launch(void* const* d_in, const int* in_sizes, int n_in,
                              void* d_out, int out_size, void* d_ws, size_t ws_size,
                              hipStream_t stream) {
  const float* emb      = (const float*)d_in[0];
  const int*   pair_ids = (const int*)d_in[1];
  const int*   dims     = (const int*)d_in[2];
  const int    npairs   = in_sizes[1] / 2;

  float* partials = (float*)d_ws;
  const int pairs_per_block = WAVES_PB * PPW;                       // 32
  const int nblocks = (npairs + pairs_per_block - 1) / pairs_per_block;  // 3125

  pair_loss_kernel<<<nblocks, BLOCK, 0, stream>>>(emb, pair_ids, dims, partials, npairs);
  final_reduce_kernel<<<1, 256, 0, stream>>>(partials, nblocks, (float*)d_out,
                                             1.0f / (float)npairs);
}